// RGCNLayer_18966575579799
// MI455X (gfx1250) — compile-verified
//
#include <hip/hip_runtime.h>

typedef __attribute__((ext_vector_type(2))) float v2f;
typedef __attribute__((ext_vector_type(4))) float v4f;
typedef __attribute__((ext_vector_type(8))) float v8f;

#define IN_F  32
#define OUT_F 32
#define RR    16
#define BB    8
#define WSZ   (RR * IN_F * OUT_F)   // 16384 floats = 64KB

// ---------------------------------------------------------------------------
// Kernel 1: basis combine, replicating torch/jax flat-reshape semantics:
//   weight [B,IN,OUT] viewed as [IN,B,OUT]; w_comp[R,B] @ view -> [IN,R,OUT];
//   reshape -> [R,IN,OUT].  For flat output idx:
//   x = idx/512, r = (idx/32)%16, z = idx%32
//   W[idx] = sum_b w_comp[r][b] * weight_flat[x*256 + b*32 + z]
// ---------------------------------------------------------------------------
__global__ void rgcn_basis_w(const float* __restrict__ weight,
                             const float* __restrict__ w_comp,
                             float* __restrict__ W) {
    int idx = blockIdx.x * blockDim.x + threadIdx.x;
    if (idx >= WSZ) return;
    int x = idx >> 9;          // idx / 512
    int r = (idx >> 5) & 15;   // (idx / 32) % 16
    int z = idx & 31;
    float acc = 0.0f;
#pragma unroll
    for (int b = 0; b < BB; ++b)
        acc = fmaf(w_comp[r * BB + b], weight[x * (BB * OUT_F) + b * OUT_F + z], acc);
    W[idx] = acc;
}

// ---------------------------------------------------------------------------
// Kernel 2: h_rel[n][r][o] = inputs[n][:] @ W[r][:][o], fp32 WMMA 16x16x4.
// One wave per 16-node tile; 4 waves / block share one LDS copy of W (64KB).
//
// A (16x4 f32) lane layout (ISA 7.12.2): lanes 0-15 hold M=lane, K={k0,k0+1};
// lanes 16-31 hold M=lane-16, K={k0+2,k0+3}.  B (4x16) symmetric:
// lanes 0-15: N=lane, rows {k0,k0+1}; lanes 16-31: N=lane-16, rows {k0+2,k0+3}.
// C/D (16x16 f32): VGPR v, lane L -> (M = v + 8*(L/16), N = L%16).
// ---------------------------------------------------------------------------
__global__ __launch_bounds__(128)
void rgcn_hrel_wmma(const float* __restrict__ x,
                    const float* __restrict__ W,
                    float* __restrict__ hrel, int n) {
    __shared__ float sW[WSZ];
    for (int i = threadIdx.x; i < WSZ; i += blockDim.x)
        sW[i] = W[i];
    __syncthreads();

    const int wave = threadIdx.x >> 5;
    const int lane = threadIdx.x & 31;
    const int tile = blockIdx.x * 4 + wave;
    const int numTiles = (n + 15) >> 4;
    if (tile >= numTiles) return;

    const int  mrow     = lane & 15;          // row within tile for A / col for B
    const bool hiHalf   = lane >= 16;
    const int  tileBase = tile << 4;
    const int  node     = tileBase + mrow;
    const bool fullTile = (tileBase + 16) <= n;   // wave-uniform

    // A operand: this lane's node row, 8 chunks of float4 (chunk s = K 4s..4s+3)
    v4f a[8] = {};
    if (node < n) {
        const float* xr = x + (size_t)node * IN_F;
#pragma unroll
        for (int c = 0; c < 8; ++c)
            a[c] = *(const v4f*)(xr + 4 * c);
    }

    const int mbase = tileBase + (hiHalf ? 8 : 0);   // D-store node base for this lane

    for (int r = 0; r < RR; ++r) {
#pragma unroll
        for (int t = 0; t < 2; ++t) {         // two 16-col output tiles
            v8f acc = {};
            const int col = mrow + 16 * t;
#pragma unroll
            for (int s = 0; s < 8; ++s) {     // K = 32 in steps of 4
                v2f av, bv;
                av.x = hiHalf ? a[s].z : a[s].x;
                av.y = hiHalf ? a[s].w : a[s].y;
                const int krow = 4 * s + (hiHalf ? 2 : 0);
                const float* wb = &sW[(r * IN_F + krow) * OUT_F + col];
                bv.x = wb[0];
                bv.y = wb[OUT_F];
                acc = __builtin_amdgcn_wmma_f32_16x16x4_f32(
                    false, av, false, bv, (short)0, acc, false, false);
            }
            // store D: lane L, vgpr v -> node mbase + v, col 16t + L%16
            float* op = &hrel[((size_t)mbase * RR + r) * OUT_F + col];
            if (fullTile) {                   // uniform branch: no EXEC juggling
#pragma unroll
                for (int v = 0; v < 8; ++v)
                    op[(size_t)v * (RR * OUT_F)] = acc[v];
            } else {
#pragma unroll
                for (int v = 0; v < 8; ++v)
                    if (mbase + v < n)
                        op[(size_t)v * (RR * OUT_F)] = acc[v];
            }
        }
    }
}

// ---------------------------------------------------------------------------
// Kernel 3: per-edge gather + scatter-sum.  8 edges per 256-thread block;
// 32 lanes = 32 output features of one edge.
// ---------------------------------------------------------------------------
__global__ void rgcn_scatter(const float* __restrict__ hrel,
                             const float* __restrict__ norm,
                             const int* __restrict__ src,
                             const int* __restrict__ dst,
                             const int* __restrict__ rel,
                             float* __restrict__ out, int e) {
    const int lane = threadIdx.x & 31;
    const int eid  = blockIdx.x * (blockDim.x >> 5) + (threadIdx.x >> 5);
    if (eid >= e) return;
    const int   s  = src[eid];
    const int   d  = dst[eid];
    const int   r  = rel[eid];
    const float nm = norm[eid];
    const float v  = hrel[((size_t)s * RR + r) * OUT_F + lane] * nm;
    unsafeAtomicAdd(&out[(size_t)d * OUT_F + lane], v);
}

// ---------------------------------------------------------------------------
// Fallback (workspace too small for h_rel): fused per-edge dot products.
// ---------------------------------------------------------------------------
__global__ void rgcn_edge_fused(const float* __restrict__ x,
                                const float* __restrict__ W,
                                const float* __restrict__ norm,
                                const int* __restrict__ src,
                                const int* __restrict__ dst,
                                const int* __restrict__ rel,
                                float* __restrict__ out, int e) {
    const int lane = threadIdx.x & 31;
    const int eid  = blockIdx.x * (blockDim.x >> 5) + (threadIdx.x >> 5);
    if (eid >= e) return;
    const int   s  = src[eid];
    const int   d  = dst[eid];
    const int   r  = rel[eid];
    const float nm = norm[eid];
    const float* xr = x + (size_t)s * IN_F;
    const float* wr = W + r * IN_F * OUT_F + lane;   // column `lane`
    float acc = 0.0f;
#pragma unroll
    for (int i = 0; i < IN_F; ++i)
        acc = fmaf(xr[i], wr[i * OUT_F], acc);
    unsafeAtomicAdd(&out[(size_t)d * OUT_F + lane], acc * nm);
}

__global__ void zero_f32(float* __restrict__ p, int n) {
    int i = blockIdx.x * blockDim.x + threadIdx.x;
    if (i < n) p[i] = 0.0f;
}

// ---------------------------------------------------------------------------
extern "C" void kernel_launch(void* const* d_in, const int* in_sizes, int n_in,
                              void* d_out, int out_size, void* d_ws, size_t ws_size,
                              hipStream_t stream) {
    const float* inputs = (const float*)d_in[0];   // [N, 32]
    const float* weight = (const float*)d_in[1];   // [8, 32, 32]
    const float* w_comp = (const float*)d_in[2];   // [16, 8]
    const float* norm   = (const float*)d_in[3];   // [E, 1]
    const int*   src    = (const int*)d_in[4];     // [E]
    const int*   dst    = (const int*)d_in[5];     // [E]
    const int*   rel    = (const int*)d_in[6];     // [E]
    float*       out    = (float*)d_out;           // [N, 32]

    const int n = in_sizes[0] / IN_F;
    const int e = in_sizes[4];

    float* W    = (float*)d_ws;                    // 64 KB
    float* hrel = W + WSZ;                         // [N, 16, 32]
    const size_t need = (size_t)WSZ * 4 + (size_t)n * RR * OUT_F * 4;

    // zero the (poisoned) output accumulator
    zero_f32<<<(n * OUT_F + 255) / 256, 256, 0, stream>>>(out, n * OUT_F);

    // basis combine
    rgcn_basis_w<<<(WSZ + 255) / 256, 256, 0, stream>>>(weight, w_comp, W);

    if (ws_size >= need) {
        const int numTiles = (n + 15) >> 4;
        rgcn_hrel_wmma<<<(numTiles + 3) / 4, 128, 0, stream>>>(inputs, W, hrel, n);
        rgcn_scatter<<<(e + 7) / 8, 256, 0, stream>>>(hrel, norm, src, dst, rel, out, e);
    } else {
        rgcn_edge_fused<<<(e + 7) / 8, 256, 0, stream>>>(inputs, W, norm, src, dst, rel, out, e);
    }
}